// Top1Gate_38319698214956
// MI455X (gfx1250) — compile-verified
//
#include <hip/hip_runtime.h>
#include <hip/hip_bf16.h>
#include <math.h>

// Problem constants (from reference)
#define T_TOK 8192
#define D_DIM 1024
#define E_EXP 64
#define CAPACITY 256
#define EPSN 1e-4f
#define TEC ((size_t)T_TOK * (size_t)E_EXP * (size_t)CAPACITY)  // 134217728

typedef float v2f __attribute__((ext_vector_type(2)));
typedef float v4f __attribute__((ext_vector_type(4)));
typedef float v8f __attribute__((ext_vector_type(8)));

// workspace layout, in floats
#define WS_MPK    0                              // packed B operand: 1024*64
#define WS_LOGITS (WS_MPK + D_DIM * E_EXP)       // 8192*64
#define WS_FMIN   (WS_LOGITS + T_TOK * E_EXP)    // 1 (ordered-uint encoded min)
#define WS_IDX    (WS_FMIN + 1)                  // 8192 int
#define WS_G1S    (WS_IDX + T_TOK)               // 8192 float (1.5 * top gate)
#define WS_BLOCKG (WS_G1S + T_TOK)               // 64 blocks * 64 experts float
#define WS_CNT    (WS_BLOCKG + 64 * E_EXP)       // 64*64 int
#define WS_OFF    (WS_CNT + 64 * E_EXP)          // 64*64 int

#define BT 128  // tokens per block for softmax/scan/scatter (T/BT = 64 chunks)

__device__ __forceinline__ unsigned enc_ordered(float x) {
  unsigned u = __float_as_uint(x);
  return (u & 0x80000000u) ? ~u : (u | 0x80000000u);
}
__device__ __forceinline__ float dec_ordered(unsigned u) {
  unsigned b = (u & 0x80000000u) ? (u ^ 0x80000000u) : ~u;
  return __uint_as_float(b);
}

// ---------------------------------------------------------------------------
// K1: fold centroid-normalization into M = W_red @ cn^T  [D, E], stored in
// the exact per-lane B-operand layout of V_WMMA_F32_16X16X4_F32 so K2's B
// loads are single contiguous b64 per lane. Also initializes the fmin cell.
// Mpk[((kstep*4 + ntile)*32 + lane)*2 + j] = M[kstep*4 + 2*(lane>>4) + j][ntile*16 + (lane&15)]
__global__ void k_pack(const float* __restrict__ Wred,
                       const float* __restrict__ cent,
                       float* __restrict__ ws) {
  int p = blockIdx.x * blockDim.x + threadIdx.x;
  if (p == 0) ((unsigned*)(ws + WS_FMIN))[0] = 0xFFFFFFFFu;  // identity for min
  if (p >= D_DIM * E_EXP) return;
  int j = p & 1;
  int rest = p >> 1;
  int lane = rest & 31;
  int t2 = rest >> 5;
  int ntile = t2 & 3;
  int kstep = t2 >> 2;
  int d = kstep * 4 + 2 * (lane >> 4) + j;
  int e = ntile * 16 + (lane & 15);
  float c0 = cent[e * 4 + 0], c1 = cent[e * 4 + 1];
  float c2 = cent[e * 4 + 2], c3 = cent[e * 4 + 3];
  float inv = 1.0f / fmaxf(sqrtf(c0 * c0 + c1 * c1 + c2 * c2 + c3 * c3), EPSN);
  float m = (Wred[d * 4 + 0] * c0 + Wred[d * 4 + 1] * c1 +
             Wred[d * 4 + 2] * c2 + Wred[d * 4 + 3] * c3) * inv;
  ws[WS_MPK + p] = m;
}

// ---------------------------------------------------------------------------
// K2: logits[T,E] = x @ M via V_WMMA_F32_16X16X4_F32.
// One wave per 16-token tile; 4 N-tiles (64 experts) share each A load.
__global__ __launch_bounds__(256) void k_gemm(const float* __restrict__ x,
                                              float* __restrict__ ws) {
  int lane = threadIdx.x & 31;
  int wave = (blockIdx.x * blockDim.x + threadIdx.x) >> 5;  // 0..511
  const float* Mpk = ws + WS_MPK;
  float* logits = ws + WS_LOGITS;
  int row0 = wave * 16;
  // A layout (16x4 f32): lanes 0-15 -> M=lane, K={k,k+1}; lanes 16-31 -> K={k+2,k+3}
  const float* xrow = x + (size_t)(row0 + (lane & 15)) * D_DIM + 2 * (lane >> 4);
  v8f acc0 = {}, acc1 = {}, acc2 = {}, acc3 = {};
  for (int ks = 0; ks < D_DIM / 4; ++ks) {
    v2f a = *(const v2f*)(xrow + ks * 4);
    const float* bp = Mpk + (size_t)ks * 256 + lane * 2;
    v2f b0 = *(const v2f*)(bp);
    v2f b1 = *(const v2f*)(bp + 64);
    v2f b2 = *(const v2f*)(bp + 128);
    v2f b3 = *(const v2f*)(bp + 192);
    acc0 = __builtin_amdgcn_wmma_f32_16x16x4_f32(false, a, false, b0, (short)0, acc0, false, false);
    acc1 = __builtin_amdgcn_wmma_f32_16x16x4_f32(false, a, false, b1, (short)0, acc1, false, false);
    acc2 = __builtin_amdgcn_wmma_f32_16x16x4_f32(false, a, false, b2, (short)0, acc2, false, false);
    acc3 = __builtin_amdgcn_wmma_f32_16x16x4_f32(false, a, false, b3, (short)0, acc3, false, false);
  }
  // C/D layout: lanes 0-15 -> N=lane, M=v; lanes 16-31 -> N=lane-16, M=v+8
  int n = lane & 15;
  int mbase = 8 * (lane >> 4);
  float mn = __builtin_inff();
  v8f accs[4] = {acc0, acc1, acc2, acc3};
#pragma unroll
  for (int nt = 0; nt < 4; ++nt) {
#pragma unroll
    for (int v = 0; v < 8; ++v) {
      float val = accs[nt][v];
      logits[(size_t)(row0 + mbase + v) * E_EXP + nt * 16 + n] = val;
      if (isfinite(val)) mn = fminf(mn, val);
    }
  }
  if (mn < __builtin_inff())
    atomicMin((unsigned*)(ws + WS_FMIN), enc_ordered(mn));  // order-independent
}

// ---------------------------------------------------------------------------
// K3: per-token softmax/argmax; deterministic per-chunk gate sums + counts.
__global__ __launch_bounds__(BT) void k_softmax(float* __restrict__ ws) {
  __shared__ float g[E_EXP][BT];
  __shared__ int sidx[BT];
  int t = blockIdx.x * BT + threadIdx.x;
  const float* logits = ws + WS_LOGITS;
  float fmin = dec_ordered(((unsigned*)(ws + WS_FMIN))[0]);
  float vmax = -__builtin_inff();
  int amax = 0;
  for (int e = 0; e < E_EXP; ++e) {
    float v = logits[(size_t)t * E_EXP + e];
    if (!isfinite(v)) v = fmin;
    if (v > vmax) { vmax = v; amax = e; }  // strict > keeps first max (jnp.argmax)
  }
  float s = 0.f;
  for (int e = 0; e < E_EXP; ++e) {
    float v = logits[(size_t)t * E_EXP + e];
    if (!isfinite(v)) v = fmin;
    float p = __expf(v - vmax);
    g[e][threadIdx.x] = p;
    s += p;
  }
  float inv_s = 1.0f / s;
  for (int e = 0; e < E_EXP; ++e) g[e][threadIdx.x] *= inv_s;
  sidx[threadIdx.x] = amax;
  ((int*)(ws + WS_IDX))[t] = amax;
  (ws + WS_G1S)[t] = 1.5f * inv_s;  // 1.5 * gate at argmax
  __syncthreads();
  if (threadIdx.x < E_EXP) {
    int e = threadIdx.x;
    float acc = 0.f;
    int cnt = 0;
    for (int j = 0; j < BT; ++j) {  // fixed order: deterministic
      acc += g[e][j];
      cnt += (sidx[j] == e);
    }
    (ws + WS_BLOCKG)[blockIdx.x * E_EXP + e] = acc;
    ((int*)(ws + WS_CNT))[blockIdx.x * E_EXP + e] = cnt;
  }
}

// ---------------------------------------------------------------------------
// K4: single block — per-expert exclusive prefix over chunks, expert_counts,
// gate sums, and l_aux = (E/T^2) * sum_e gsum[e]*cnt[e].
__global__ __launch_bounds__(E_EXP) void k_scan(float* __restrict__ ws,
                                                float* __restrict__ out) {
  __shared__ float part[E_EXP];
  int e = threadIdx.x;
  const int NB = T_TOK / BT;  // 64 chunks
  int run = 0;
  float gs = 0.f;
  for (int b = 0; b < NB; ++b) {
    ((int*)(ws + WS_OFF))[b * E_EXP + e] = run;
    run += ((int*)(ws + WS_CNT))[b * E_EXP + e];
    gs += (ws + WS_BLOCKG)[b * E_EXP + e];
  }
  out[1 + 2 * TEC + e] = (float)run;  // expert_counts (pre-capacity)
  part[e] = gs * (float)run;
  __syncthreads();
  if (e == 0) {
    float s = 0.f;
    for (int i = 0; i < E_EXP; ++i) s += part[i];
    out[0] = s * ((float)E_EXP / ((float)T_TOK * (float)T_TOK));  // l_aux
  }
}

// ---------------------------------------------------------------------------
// K5: streaming zero-fill of combine+dispatch (2*TEC floats at out+1).
// out+1 is only 4B aligned: 3-element head, b128 NT body from out+4, 1 tail.
__global__ void k_zero(float* __restrict__ out) {
  const size_t n4 = (2 * TEC - 4) / 4;  // 67108863 float4s
  size_t tid = (size_t)blockIdx.x * blockDim.x + threadIdx.x;
  size_t stride = (size_t)gridDim.x * blockDim.x;
  v4f z = {0.f, 0.f, 0.f, 0.f};
  v4f* base = (v4f*)(out + 4);
  for (size_t i = tid; i < n4; i += stride)
    __builtin_nontemporal_store(z, base + i);
  if (tid == 0) {
    out[1] = 0.f; out[2] = 0.f; out[3] = 0.f;
    out[2 * TEC] = 0.f;  // last element of region [1, 1+2*TEC)
  }
}

// ---------------------------------------------------------------------------
// K6: scatter the 8192 nonzeros (loc = chunk offset + in-chunk rank).
__global__ __launch_bounds__(BT) void k_scatter(const float* __restrict__ ws,
                                                float* __restrict__ out) {
  __shared__ int sidx[BT];
  int tloc = threadIdx.x;
  int t = blockIdx.x * BT + tloc;
  int e = ((const int*)(ws + WS_IDX))[t];
  sidx[tloc] = e;
  __syncthreads();
  int rank = 0;
  for (int j = 0; j < tloc; ++j) rank += (sidx[j] == e);
  int loc = ((const int*)(ws + WS_OFF))[blockIdx.x * E_EXP + e] + rank;
  if (loc < CAPACITY) {
    size_t o = ((size_t)t * E_EXP + (size_t)e) * CAPACITY + (size_t)loc;
    out[1 + o] = (ws + WS_G1S)[t];    // combine
    out[1 + TEC + o] = 1.0f;          // dispatch_mask (true -> 1.0)
  }
}

// ---------------------------------------------------------------------------
extern "C" void kernel_launch(void* const* d_in, const int* in_sizes, int n_in,
                              void* d_out, int out_size, void* d_ws, size_t ws_size,
                              hipStream_t stream) {
  const float* x = (const float*)d_in[0];         // [8192,1024]
  const float* Wred = (const float*)d_in[1];      // [1024,4]
  const float* cent = (const float*)d_in[2];      // [64,4]
  float* out = (float*)d_out;
  float* ws = (float*)d_ws;

  // K1: pack M (65536 entries) + init fmin
  k_pack<<<(D_DIM * E_EXP + 255) / 256, 256, 0, stream>>>(Wred, cent, ws);
  // K2: WMMA GEMM -> logits (512 waves, 8 waves/block)
  k_gemm<<<64, 256, 0, stream>>>(x, ws);
  // K3: softmax / argmax / chunk stats (64 chunks of 128 tokens)
  k_softmax<<<T_TOK / BT, BT, 0, stream>>>(ws);
  // K4: prefix scan + l_aux + expert_counts
  k_scan<<<1, E_EXP, 0, stream>>>(ws, out);
  // K5: zero the 1.07 GB combine+dispatch region (NT streaming stores)
  k_zero<<<4096, 256, 0, stream>>>(out);
  // K6: scatter nonzeros (after zero-fill on same stream)
  k_scatter<<<T_TOK / BT, BT, 0, stream>>>(ws, out);
}